// RelGraphConvLayer_38302518346209
// MI455X (gfx1250) — compile-verified
//
#include <hip/hip_runtime.h>

#define DD 64

typedef float v2f __attribute__((ext_vector_type(2)));
typedef float v8f __attribute__((ext_vector_type(8)));

// ---------------------------------------------------------------- utilities

__global__ __launch_bounds__(256) void zero_kernel(float* __restrict__ p, int n) {
  int i = blockIdx.x * blockDim.x + threadIdx.x;
  if (i < n) p[i] = 0.0f;
}

__global__ __launch_bounds__(256) void count_deg_kernel(const int* __restrict__ dst, int e,
                                                        float* __restrict__ deg) {
  int i = blockIdx.x * blockDim.x + threadIdx.x;
  if (i < e) unsafeAtomicAdd(deg + dst[i], 1.0f);
}

__global__ __launch_bounds__(256) void inv_deg_kernel(float* __restrict__ deg, int n) {
  int i = blockIdx.x * blockDim.x + threadIdx.x;
  if (i < n) deg[i] = 1.0f / fmaxf(deg[i], 1.0f);
}

// -------------------------------------------------- dual projection via WMMA
// One wave computes a 16-row strip of BOTH Y1 = X@W1 and Y2 = X@W2 (A reuse).
// Full fp32 precision via V_WMMA_F32_16X16X4_F32 (fp32 A/B, fp32 accum).

__global__ __launch_bounds__(256) void project_dual_kernel(
    const float* __restrict__ X, const float* __restrict__ W1,
    const float* __restrict__ W2, float* __restrict__ Y1, float* __restrict__ Y2,
    int ntiles) {
  int wid  = (int)((blockIdx.x * blockDim.x + threadIdx.x) >> 5);  // wave-uniform
  int lane = (int)(threadIdx.x & 31);
  if (wid >= ntiles) return;  // uniform per wave -> EXEC stays all-ones for WMMA

  const int m     = lane & 15;         // row within 16x16 tile (A and D col index)
  const int khalf = (lane >> 4) * 2;   // 0 for lanes 0-15, 2 for lanes 16-31

  const float* xrow = X + (size_t)(wid * 16 + m) * DD;

  v8f acc1[4] = {};  // 4 column tiles of Y1 strip
  v8f acc2[4] = {};  // 4 column tiles of Y2 strip

  for (int k0 = 0; k0 < DD; k0 += 4) {
    // A fragment: 16x4 fp32.  VGPR0 = A[m][khalf], VGPR1 = A[m][khalf+1]
    v2f a = *(const v2f*)(xrow + k0 + khalf);
#pragma unroll
    for (int t = 0; t < 4; ++t) {
      int col = t * 16 + m;
      // B fragment: 4x16 fp32.  VGPR0 = B[khalf][col], VGPR1 = B[khalf+1][col]
      v2f b1, b2;
      b1.x = W1[(k0 + khalf) * DD + col];
      b1.y = W1[(k0 + khalf + 1) * DD + col];
      b2.x = W2[(k0 + khalf) * DD + col];
      b2.y = W2[(k0 + khalf + 1) * DD + col];
      acc1[t] = __builtin_amdgcn_wmma_f32_16x16x4_f32(false, a, false, b1,
                                                      (short)0, acc1[t], false, false);
      acc2[t] = __builtin_amdgcn_wmma_f32_16x16x4_f32(false, a, false, b2,
                                                      (short)0, acc2[t], false, false);
    }
  }

  // D layout: VGPR r -> row (r + 8*(lane>=16)), col = lane&15 within tile
  const int rbase = (lane >> 4) * 8;
#pragma unroll
  for (int t = 0; t < 4; ++t) {
    int col = t * 16 + m;
#pragma unroll
    for (int r = 0; r < 8; ++r) {
      size_t off = (size_t)(wid * 16 + rbase + r) * DD + col;
      Y1[off] = acc1[t][r];
      Y2[off] = acc2[t][r];
    }
  }
}

// ------------------------------------------------------ edge gather/scatter
// One wave per edge: 32 lanes x float2 = 64 features. Pre-scaled by 1/deg(dst)
// so both relations per dst type accumulate directly into the shared output.

__global__ __launch_bounds__(256) void scatter_edges_kernel(
    const float* __restrict__ Y, const int* __restrict__ src,
    const int* __restrict__ dst, const float* __restrict__ inv,
    float* __restrict__ out, int e) {
  int wid  = (int)((blockIdx.x * blockDim.x + threadIdx.x) >> 5);
  int lane = (int)(threadIdx.x & 31);
  if (wid >= e) return;
  int s = src[wid];
  int d = dst[wid];
  float scale = inv[d];
  v2f v = *(const v2f*)(Y + (size_t)s * DD + lane * 2);
  float* o = out + (size_t)d * DD + lane * 2;
  unsafeAtomicAdd(o,     v.x * scale);
  unsafeAtomicAdd(o + 1, v.y * scale);
}

// ------------------------------------------------------------ bias + ReLU

__global__ __launch_bounds__(256) void finalize_kernel(float* __restrict__ out,
                                                       const float* __restrict__ bias,
                                                       int n) {
  int i = blockIdx.x * blockDim.x + threadIdx.x;
  if (i < n) out[i] = fmaxf(out[i] + bias[i & (DD - 1)], 0.0f);
}

// ----------------------------------------------------------------- launcher

extern "C" void kernel_launch(void* const* d_in, const int* in_sizes, int n_in,
                              void* d_out, int out_size, void* d_ws, size_t ws_size,
                              hipStream_t stream) {
  const float* x_drug = (const float*)d_in[0];
  const float* x_gene = (const float*)d_in[1];
  const float* W_dd   = (const float*)d_in[2];
  const float* W_dg   = (const float*)d_in[3];
  const float* W_gd   = (const float*)d_in[4];
  const float* W_gg   = (const float*)d_in[5];
  const float* bias   = (const float*)d_in[6];
  const int* src_dd = (const int*)d_in[7];
  const int* dst_dd = (const int*)d_in[8];
  const int* src_dg = (const int*)d_in[9];
  const int* dst_dg = (const int*)d_in[10];
  const int* src_gd = (const int*)d_in[11];
  const int* dst_gd = (const int*)d_in[12];
  const int* src_gg = (const int*)d_in[13];
  const int* dst_gg = (const int*)d_in[14];

  const int n = in_sizes[0] / DD;  // 100000 nodes per type
  const int e = in_sizes[7];       // 800000 edges per relation

  float* out      = (float*)d_out;
  float* out_drug = out;
  float* out_gene = out + (size_t)n * DD;

  // workspace layout: yA | yB | inv_dd | inv_dg | inv_gd | inv_gg
  float* yA     = (float*)d_ws;
  float* yB     = yA + (size_t)n * DD;
  float* inv_dd = yB + (size_t)n * DD;
  float* inv_dg = inv_dd + n;
  float* inv_gd = inv_dg + n;
  float* inv_gg = inv_gd + n;

  const int B = 256;
  auto cdiv = [](int a, int b) { return (a + b - 1) / b; };

  // 1) zero output accumulator and degree arrays
  zero_kernel<<<cdiv(2 * n * DD, B), B, 0, stream>>>(out, 2 * n * DD);
  zero_kernel<<<cdiv(4 * n, B), B, 0, stream>>>(inv_dd, 4 * n);

  // 2) in-degrees per relation, then invert (1/max(deg,1))
  count_deg_kernel<<<cdiv(e, B), B, 0, stream>>>(dst_dd, e, inv_dd);
  count_deg_kernel<<<cdiv(e, B), B, 0, stream>>>(dst_dg, e, inv_dg);
  count_deg_kernel<<<cdiv(e, B), B, 0, stream>>>(dst_gd, e, inv_gd);
  count_deg_kernel<<<cdiv(e, B), B, 0, stream>>>(dst_gg, e, inv_gg);
  inv_deg_kernel<<<cdiv(4 * n, B), B, 0, stream>>>(inv_dd, 4 * n);

  const int ntiles  = n / 16;                 // 6250 (N divisible by 16)
  const int pblocks = cdiv(ntiles * 32, B);
  const int eblocks = cdiv(e, 8);             // one wave per edge, 8 waves/block

  // 3) drug-sourced projections, then scatter (stream order serializes reuse)
  project_dual_kernel<<<pblocks, B, 0, stream>>>(x_drug, W_dd, W_dg, yA, yB, ntiles);
  scatter_edges_kernel<<<eblocks, B, 0, stream>>>(yA, src_dd, dst_dd, inv_dd, out_drug, e);
  scatter_edges_kernel<<<eblocks, B, 0, stream>>>(yB, src_dg, dst_dg, inv_dg, out_gene, e);

  // 4) gene-sourced projections reuse yA/yB, then scatter
  project_dual_kernel<<<pblocks, B, 0, stream>>>(x_gene, W_gd, W_gg, yA, yB, ntiles);
  scatter_edges_kernel<<<eblocks, B, 0, stream>>>(yA, src_gd, dst_gd, inv_gd, out_drug, e);
  scatter_edges_kernel<<<eblocks, B, 0, stream>>>(yB, src_gg, dst_gg, inv_gg, out_gene, e);

  // 5) bias + ReLU in place
  finalize_kernel<<<cdiv(2 * n * DD, B), B, 0, stream>>>(out, bias, 2 * n * DD);
}